// BahdanauAttnDecoderRNN_45329084841990
// MI455X (gfx1250) — compile-verified
//
#include <hip/hip_runtime.h>
#include <math.h>

// Problem sizes (match reference)
constexpr int kH = 256;   // hidden
constexpr int kV = 62;    // vocab
constexpr int kS = 512;   // encoder seq
constexpr int kB = 64;    // batch
constexpr int kT = 32;    // decode steps

typedef float v2f __attribute__((ext_vector_type(2)));
typedef float v8f __attribute__((ext_vector_type(8)));

// ---------------------------------------------------------------------------
// WMMA f32 16x16x4 fragment helpers (wave32, one wave per block for GEMM kernels)
//
// A fragment (16x4, row-major source, leading dim ld):
//   lanes 0-15 : M = lane,    VGPR0 = K=k0+0, VGPR1 = K=k0+1
//   lanes 16-31: M = lane-16, VGPR0 = K=k0+2, VGPR1 = K=k0+3
// For C = A @ W^T, the B fragment B[k][n] = W[n][k] is loaded with the *same*
// pattern, using W's row index n in place of M.
// ---------------------------------------------------------------------------
__device__ __forceinline__ v2f load_frag(const float* __restrict__ base,
                                         int row0, int ld, int k0) {
    int lane = threadIdx.x & 31;
    int l16  = lane & 15;
    int half = lane >> 4;
    const float* p = base + (size_t)(row0 + l16) * ld + k0 + 2 * half;
    v2f f; f.x = p[0]; f.y = p[1];
    return f;
}

__device__ __forceinline__ v2f load_frag_rowclamp(const float* __restrict__ base,
                                                  int row0, int rowmax, int ld, int k0) {
    int lane = threadIdx.x & 31;
    int l16  = lane & 15;
    int half = lane >> 4;
    int r = row0 + l16;
    v2f f;
    if (r < rowmax) {
        const float* p = base + (size_t)r * ld + k0 + 2 * half;
        f.x = p[0]; f.y = p[1];
    } else {
        f.x = 0.0f; f.y = 0.0f;
    }
    return f;
}

__device__ __forceinline__ v8f wmma_f32_k4(v2f a, v2f b, v8f c) {
    return __builtin_amdgcn_wmma_f32_16x16x4_f32(false, a, false, b,
                                                 (short)0, c, false, false);
}

__device__ __forceinline__ float sigmoidf_(float x) {
    return 1.0f / (1.0f + __expf(-x));
}

// ---------------------------------------------------------------------------
// h := 0
// ---------------------------------------------------------------------------
__global__ void zero_kernel(float* __restrict__ p, int n) {
    int i = blockIdx.x * blockDim.x + threadIdx.x;
    if (i < n) p[i] = 0.0f;
}

// ---------------------------------------------------------------------------
// Embedding gather: xin[b, 0:H] = emb[target_seq[t, b]]
// grid = B blocks, H threads
// ---------------------------------------------------------------------------
__global__ void embed_kernel(const int* __restrict__ tseq,
                             const float* __restrict__ emb,
                             float* __restrict__ xin, int t) {
    int b = blockIdx.x;
    int h = threadIdx.x;
    int tok = tseq[t * kB + b];
    xin[b * (2 * kH) + h] = emb[tok * kH + h];
}

// ---------------------------------------------------------------------------
// Fused additive attention: scores -> softmax(S) -> context.
// One block per batch element b; 256 threads = 8 waves.
// Score phase: wave w handles s = w, w+8, ... ; lanes stripe over H.
// Context phase: thread h accumulates over S (encoder slab is L2-resident).
// Writes ctx into xin[b, H:2H].
// ---------------------------------------------------------------------------
__global__ void attn_ctx_kernel(const float* __restrict__ enc,
                                const float* __restrict__ hst,
                                const float* __restrict__ v_w,
                                const float* __restrict__ v_b,
                                float* __restrict__ xin) {
    __shared__ float sc[kS];
    __shared__ float red[256];

    int b    = blockIdx.x;
    int tid  = threadIdx.x;
    int lane = tid & 31;
    int wave = tid >> 5;

    // Hoist per-lane h-state and v weights (fixed across s)
    float hv[8], vw[8];
    #pragma unroll
    for (int i = 0; i < 8; ++i) {
        int h  = lane + 32 * i;
        hv[i]  = hst[b * kH + h];
        vw[i]  = v_w[h];
    }

    // scores[s] = sum_h v_w[h] * tanh(h[b,h] + enc[s,b,h]) + v_b
    for (int s = wave; s < kS; s += 8) {
        const float* ep = enc + (size_t)(s * kB + b) * kH;
        float partial = 0.0f;
        #pragma unroll
        for (int i = 0; i < 8; ++i) {
            int h = lane + 32 * i;
            partial += vw[i] * tanhf(hv[i] + ep[h]);
        }
        #pragma unroll
        for (int off = 16; off > 0; off >>= 1)
            partial += __shfl_xor(partial, off, 32);
        if (lane == 0) sc[s] = partial + v_b[0];
    }
    __syncthreads();

    // softmax over S=512 with 256 threads
    float m = fmaxf(sc[tid], sc[tid + 256]);
    red[tid] = m;
    __syncthreads();
    for (int off = 128; off > 0; off >>= 1) {
        if (tid < off) red[tid] = fmaxf(red[tid], red[tid + off]);
        __syncthreads();
    }
    m = red[0];
    __syncthreads();
    float e0 = __expf(sc[tid] - m);
    float e1 = __expf(sc[tid + 256] - m);
    sc[tid] = e0;
    sc[tid + 256] = e1;
    red[tid] = e0 + e1;
    __syncthreads();
    for (int off = 128; off > 0; off >>= 1) {
        if (tid < off) red[tid] += red[tid + off];
        __syncthreads();
    }
    float inv = 1.0f / red[0];

    // ctx[b, h=tid] = sum_s a[s] * enc[s,b,h]
    float acc = 0.0f;
    for (int s = 0; s < kS; ++s) {
        const float* ep = enc + (size_t)(s * kB + b) * kH;
        if (s + 8 < kS)
            __builtin_prefetch(enc + (size_t)((s + 8) * kB + b) * kH + tid, 0, 0);
        acc += sc[s] * ep[tid];
    }
    xin[b * (2 * kH) + kH + tid] = acc * inv;
}

// ---------------------------------------------------------------------------
// x = relu(xin @ wc^T + bc)   [64 x 512] @ [512 x 256]
// grid = (B/16, H/16), 32 threads (one wave) per 16x16 output tile.
// ---------------------------------------------------------------------------
__global__ void combine_kernel(const float* __restrict__ xin,
                               const float* __restrict__ wc,
                               const float* __restrict__ bc,
                               float* __restrict__ x) {
    int m0 = blockIdx.x * 16;   // batch tile
    int n0 = blockIdx.y * 16;   // hidden tile
    v8f c = {};
    for (int k0 = 0; k0 < 2 * kH; k0 += 4) {
        v2f a  = load_frag(xin, m0, 2 * kH, k0);
        v2f bf = load_frag(wc,  n0, 2 * kH, k0);   // wc rows are output cols
        c = wmma_f32_k4(a, bf, c);
    }
    int lane = threadIdx.x & 31;
    int l16  = lane & 15;
    int half = lane >> 4;
    int col  = n0 + l16;
    float bias = bc[col];
    #pragma unroll
    for (int i = 0; i < 8; ++i) {
        int row = m0 + i + 8 * half;
        x[row * kH + col] = fmaxf(c[i] + bias, 0.0f);
    }
}

// ---------------------------------------------------------------------------
// Fused GRU cell: computes all six gate GEMM tiles with WMMA, then the
// elementwise gate math, writing h_new. grid = (B/16, H/16), one wave.
// w_ih/w_hh are [3H, H], gate order r,z,n (rows 0..H-1, H..2H-1, 2H..3H-1).
// ---------------------------------------------------------------------------
__global__ void gru_kernel(const float* __restrict__ x,
                           const float* __restrict__ hold,
                           const float* __restrict__ w_ih,
                           const float* __restrict__ w_hh,
                           const float* __restrict__ b_ih,
                           const float* __restrict__ b_hh,
                           float* __restrict__ hnew) {
    int m0 = blockIdx.x * 16;   // batch tile
    int n0 = blockIdx.y * 16;   // hidden tile
    v8f cir = {}, ciz = {}, cin = {};
    v8f chr_ = {}, chz = {}, chn = {};
    for (int k0 = 0; k0 < kH; k0 += 4) {
        v2f ax = load_frag(x,    m0, kH, k0);
        v2f ah = load_frag(hold, m0, kH, k0);
        v2f br = load_frag(w_ih, n0,          kH, k0);
        v2f bz = load_frag(w_ih, n0 + kH,     kH, k0);
        v2f bn = load_frag(w_ih, n0 + 2 * kH, kH, k0);
        cir = wmma_f32_k4(ax, br, cir);
        ciz = wmma_f32_k4(ax, bz, ciz);
        cin = wmma_f32_k4(ax, bn, cin);
        v2f hr = load_frag(w_hh, n0,          kH, k0);
        v2f hz = load_frag(w_hh, n0 + kH,     kH, k0);
        v2f hn = load_frag(w_hh, n0 + 2 * kH, kH, k0);
        chr_ = wmma_f32_k4(ah, hr, chr_);
        chz  = wmma_f32_k4(ah, hz, chz);
        chn  = wmma_f32_k4(ah, hn, chn);
    }
    int lane = threadIdx.x & 31;
    int l16  = lane & 15;
    int half = lane >> 4;
    int j = n0 + l16;
    float bir = b_ih[j], biz = b_ih[kH + j], bin = b_ih[2 * kH + j];
    float bhr = b_hh[j], bhz = b_hh[kH + j], bhn = b_hh[2 * kH + j];
    #pragma unroll
    for (int i = 0; i < 8; ++i) {
        int row = m0 + i + 8 * half;
        float r = sigmoidf_((cir[i] + bir) + (chr_[i] + bhr));
        float z = sigmoidf_((ciz[i] + biz) + (chz[i] + bhz));
        float n = tanhf((cin[i] + bin) + r * (chn[i] + bhn));
        float hp = hold[row * kH + j];
        hnew[row * kH + j] = (1.0f - z) * n + z * hp;
    }
}

// ---------------------------------------------------------------------------
// logits: out[b, t, :] = h_new @ w_out^T + b_out   (V=62 -> masked edge tile)
// grid = (B/16, ceil(V/16)), one wave per tile.
// ---------------------------------------------------------------------------
__global__ void logits_kernel(const float* __restrict__ hn,
                              const float* __restrict__ w_out,
                              const float* __restrict__ b_out,
                              float* __restrict__ out, int t) {
    int m0 = blockIdx.x * 16;
    int n0 = blockIdx.y * 16;
    v8f c = {};
    for (int k0 = 0; k0 < kH; k0 += 4) {
        v2f a  = load_frag(hn, m0, kH, k0);
        v2f bf = load_frag_rowclamp(w_out, n0, kV, kH, k0);
        c = wmma_f32_k4(a, bf, c);
    }
    int lane = threadIdx.x & 31;
    int l16  = lane & 15;
    int half = lane >> 4;
    int col  = n0 + l16;
    if (col < kV) {
        float bias = b_out[col];
        #pragma unroll
        for (int i = 0; i < 8; ++i) {
            int row = m0 + i + 8 * half;   // batch index
            out[(size_t)row * kT * kV + t * kV + col] = c[i] + bias;
        }
    }
}

// ---------------------------------------------------------------------------
// Host driver: 32 sequential steps, 5 kernels per step, double-buffered h.
// ---------------------------------------------------------------------------
extern "C" void kernel_launch(void* const* d_in, const int* in_sizes, int n_in,
                              void* d_out, int out_size, void* d_ws, size_t ws_size,
                              hipStream_t stream) {
    const float* enc   = (const float*)d_in[0];   // [S,B,H]
    const int*   tseq  = (const int*)  d_in[1];   // [T,B]
    const float* emb   = (const float*)d_in[2];   // [V,H]
    const float* v_w   = (const float*)d_in[3];   // [H]
    const float* v_b   = (const float*)d_in[4];   // scalar
    const float* wc    = (const float*)d_in[5];   // [H,2H]
    const float* bc    = (const float*)d_in[6];   // [H]
    const float* w_ih  = (const float*)d_in[7];   // [3H,H]
    const float* w_hh  = (const float*)d_in[8];   // [3H,H]
    const float* b_ih  = (const float*)d_in[9];   // [3H]
    const float* b_hh  = (const float*)d_in[10];  // [3H]
    const float* w_out = (const float*)d_in[11];  // [V,H]
    const float* b_out = (const float*)d_in[12];  // [V]
    float* out = (float*)d_out;                   // [B,T,V]

    float* ws     = (float*)d_ws;
    float* h_ping = ws;                             // [B,H]
    float* h_pong = ws + kB * kH;                   // [B,H]
    float* xin    = ws + 2 * kB * kH;               // [B,2H]  (emb | ctx)
    float* xbuf   = ws + 2 * kB * kH + kB * 2 * kH; // [B,H]

    zero_kernel<<<(kB * kH + 255) / 256, 256, 0, stream>>>(h_ping, kB * kH);

    float* hcur  = h_ping;
    float* hnext = h_pong;
    for (int t = 0; t < kT; ++t) {
        embed_kernel<<<kB, kH, 0, stream>>>(tseq, emb, xin, t);
        attn_ctx_kernel<<<kB, 256, 0, stream>>>(enc, hcur, v_w, v_b, xin);
        combine_kernel<<<dim3(kB / 16, kH / 16), 32, 0, stream>>>(xin, wc, bc, xbuf);
        gru_kernel<<<dim3(kB / 16, kH / 16), 32, 0, stream>>>(xbuf, hcur, w_ih, w_hh,
                                                              b_ih, b_hh, hnext);
        logits_kernel<<<dim3(kB / 16, (kV + 15) / 16), 32, 0, stream>>>(hnext, w_out,
                                                                        b_out, out, t);
        float* tmp = hcur; hcur = hnext; hnext = tmp;
    }
}